// EPropLSTM_67156108640263
// MI455X (gfx1250) — compile-verified
//
#include <hip/hip_runtime.h>

// e-prop LSTM for MI455X (gfx1250, wave32).
// Phase 1: sequential LSTM with V_WMMA_F32_16X16X4_F32 gate GEMM (exact fp32).
// Phase 2: single streaming kernel materializes eligibility traces with ev
//          held in registers across all T steps; NT stores (write-once).

typedef float v2f __attribute__((ext_vector_type(2)));
typedef float v8f __attribute__((ext_vector_type(8)));
typedef float f4  __attribute__((ext_vector_type(4)));

#define T_ 100
#define B_ 64
#define I_ 256
#define H_ 512
#define G4_ (4 * H_)   // 2048
#define G3_ (3 * H_)   // 1536

// ---------------------------------------------------------------- init c ----
__global__ void init_c_kernel(const float* __restrict__ c0, float* __restrict__ c) {
  int i = blockIdx.x * 256 + threadIdx.x;
  if (i < B_ * H_) c[i] = c0[i];
}

// -------------------------------------------------------------- gate GEMM ----
// gates[b,n] = bias_ih[n]+bias_hh[n] + x_t[b,:]·Wih[n,:] + h_{t-1}[b,:]·Whh[n,:]
// One 16x16 output tile per wave; 4 M-tiles x 128 N-tiles = 512 waves.
// A lane layout (16x4 f32): M = lane&15, K pair = (lane>>4)*2 .. +1
// B lane layout (4x16 f32): N = lane&15, K pair = (lane>>4)*2 .. +1
// C lane layout: N = lane&15, M = vgpr + (lane>>4)*8
__global__ __launch_bounds__(256) void gates_gemm_kernel(
    const float* __restrict__ X,    // [T,B,I]
    const float* __restrict__ H0,   // [B,H]
    const float* __restrict__ OUT,  // [T,B,H] (h history)
    const float* __restrict__ Wih,  // [4H,I]
    const float* __restrict__ Whh,  // [4H,H]
    const float* __restrict__ bih,
    const float* __restrict__ bhh,
    float* __restrict__ gates,      // [B,4H]
    int t) {
  int wave  = (blockIdx.x * 256 + (int)threadIdx.x) >> 5;
  int lane  = threadIdx.x & 31;
  int tileN = wave & 127;
  int tileM = wave >> 7;           // 0..3
  int r     = lane & 15;
  int kh    = (lane >> 4) << 1;    // 0 or 2
  int n     = tileN * 16 + r;

  float bias = bih[n] + bhh[n];
  v8f acc0, acc1;
#pragma unroll
  for (int v = 0; v < 8; ++v) { acc0[v] = bias; acc1[v] = 0.0f; }

  // --- x_t @ Wih^T, K = 256 (split across two accumulators to cut RAW chain)
  const float* xrow = X + ((size_t)t * B_ + (size_t)(tileM * 16 + r)) * I_;
  const float* wr0  = Wih + (size_t)n * I_;
  for (int k = 0; k < I_; k += 8) {
    v2f a0 = *(const v2f*)(xrow + k + kh);
    v2f b0 = *(const v2f*)(wr0  + k + kh);
    acc0 = __builtin_amdgcn_wmma_f32_16x16x4_f32(false, a0, false, b0,
                                                 (short)0, acc0, false, false);
    v2f a1 = *(const v2f*)(xrow + k + 4 + kh);
    v2f b1 = *(const v2f*)(wr0  + k + 4 + kh);
    acc1 = __builtin_amdgcn_wmma_f32_16x16x4_f32(false, a1, false, b1,
                                                 (short)0, acc1, false, false);
  }
  // --- h_{t-1} @ Whh^T, K = 512
  const float* hrow = (t == 0)
      ? (H0 + (size_t)(tileM * 16 + r) * H_)
      : (OUT + ((size_t)(t - 1) * B_ + (size_t)(tileM * 16 + r)) * H_);
  const float* wr1 = Whh + (size_t)n * H_;
  for (int k = 0; k < H_; k += 8) {
    v2f a0 = *(const v2f*)(hrow + k + kh);
    v2f b0 = *(const v2f*)(wr1  + k + kh);
    acc0 = __builtin_amdgcn_wmma_f32_16x16x4_f32(false, a0, false, b0,
                                                 (short)0, acc0, false, false);
    v2f a1 = *(const v2f*)(hrow + k + 4 + kh);
    v2f b1 = *(const v2f*)(wr1  + k + 4 + kh);
    acc1 = __builtin_amdgcn_wmma_f32_16x16x4_f32(false, a1, false, b1,
                                                 (short)0, acc1, false, false);
  }

  int mb = tileM * 16 + ((lane >> 4) << 3);
  float* gp = gates + (size_t)mb * G4_ + (size_t)tileN * 16 + r;
#pragma unroll
  for (int v = 0; v < 8; ++v) gp[(size_t)v * G4_] = acc0[v] + acc1[v];
}

// -------------------------------------------------------------- pointwise ----
__global__ __launch_bounds__(256) void pointwise_kernel(
    const float* __restrict__ gates, float* __restrict__ c,
    float* __restrict__ OUT, float* __restrict__ CX,
    float* __restrict__ f_all,   // [T,B,H]
    float* __restrict__ d_all,   // [T,B,3H]
    int t) {
  int idx = blockIdx.x * 256 + threadIdx.x;   // [0, B*H)
  int b = idx >> 9;
  int n = idx & (H_ - 1);
  const float* g = gates + (size_t)b * G4_;
  float ig = g[n], fg = g[H_ + n], gg = g[2 * H_ + n], og = g[3 * H_ + n];
  float i  = 1.0f / (1.0f + __expf(-ig));
  float f  = 1.0f / (1.0f + __expf(-fg));
  float gt = tanhf(gg);
  float o  = 1.0f / (1.0f + __expf(-og));
  float cx = c[idx];
  float cy = f * cx + i * gt;
  float hy = o * tanhf(cy);
  c[idx]  = cy;
  CX[idx] = cy;                       // final step leaves correct cx
  OUT[(size_t)t * B_ * H_ + idx] = hy;
  f_all[(size_t)t * B_ * H_ + idx] = f;
  float* d = d_all + ((size_t)t * B_ + b) * G3_ + n;
  d[0]      = gt * i * (1.0f - i);
  d[H_]     = cx * f * (1.0f - f);
  d[2 * H_] = i  * (1.0f - gt * gt);
}

// ----------------------------------------------------- phase-2 trace kernel ----
// Block = 192 threads = one (x-row + h-row) of 768 floats as float4s, covering
// 4 consecutive (b,j) rows. ev kept in registers across all T steps; f/d in LDS.
__global__ __launch_bounds__(192) void trace_kernel(
    const float* __restrict__ X,      // [T,B,I]
    const float* __restrict__ H0,     // [B,H]
    const float* __restrict__ OUT,    // [T,B,H]
    const float* __restrict__ f_all,  // [T,B,H]
    const float* __restrict__ d_all,  // [T,B,3H]
    float* __restrict__ EVIH,         // [B,3H,I]
    float* __restrict__ EVHH,         // [B,3H,H]
    float* __restrict__ EVB) {        // [B,3H]
  __shared__ float sF[4 * T_];
  __shared__ float sD[4 * T_];
  int tid  = threadIdx.x;
  int row0 = blockIdx.x * 4;          // = b*3H + j0
  int b    = row0 / G3_;
  int j0   = row0 - b * G3_;

  for (int u = tid; u < 4 * T_; u += 192) {
    int jj = u / T_;
    int t  = u - jj * T_;
    sF[u] = f_all[((size_t)t * B_ + b) * H_ + ((j0 + jj) & (H_ - 1))];
    sD[u] = d_all[((size_t)t * B_ + b) * G3_ + (j0 + jj)];
  }
  __syncthreads();

  bool isIH = tid < 64;                       // wave-aligned split (2 + 4 waves)
  int  kk   = isIH ? tid : (tid - 64);

  f4 ev0, ev1, ev2, ev3;
  // t = 0 (ev starts at zero): ev = d0 * x0
  {
    const float* p0 = isIH ? (X + (size_t)b * I_ + (size_t)kk * 4)
                           : (H0 + (size_t)b * H_ + (size_t)kk * 4);
    f4 xv = *(const f4*)p0;
    ev0 = sD[0 * T_] * xv;
    ev1 = sD[1 * T_] * xv;
    ev2 = sD[2 * T_] * xv;
    ev3 = sD[3 * T_] * xv;
  }
  const float* p;
  size_t stride;
  if (isIH) { p = X   + ((size_t)1 * B_ + b) * I_ + (size_t)kk * 4; stride = (size_t)B_ * I_; }
  else      { p = OUT + ((size_t)b) * H_          + (size_t)kk * 4; stride = (size_t)B_ * H_; }

  for (int t = 1; t < T_; ++t) {
    f4 xv = *(const f4*)p;               // L2-resident, reused 1536x per b
    p += stride;
    float fa = sF[0 * T_ + t], da = sD[0 * T_ + t];
    float fb = sF[1 * T_ + t], db = sD[1 * T_ + t];
    float fc = sF[2 * T_ + t], dc = sD[2 * T_ + t];
    float fd = sF[3 * T_ + t], dd = sD[3 * T_ + t];
    ev0 = fa * ev0 + da * xv;
    ev1 = fb * ev1 + db * xv;
    ev2 = fc * ev2 + dc * xv;
    ev3 = fd * ev3 + dd * xv;
  }

  if (isIH) {
    float* e = EVIH + (size_t)row0 * I_ + (size_t)kk * 4;
    __builtin_nontemporal_store(ev0, (f4*)(e));
    __builtin_nontemporal_store(ev1, (f4*)(e + I_));
    __builtin_nontemporal_store(ev2, (f4*)(e + 2 * I_));
    __builtin_nontemporal_store(ev3, (f4*)(e + 3 * I_));
  } else {
    float* e = EVHH + (size_t)row0 * H_ + (size_t)kk * 4;
    __builtin_nontemporal_store(ev0, (f4*)(e));
    __builtin_nontemporal_store(ev1, (f4*)(e + H_));
    __builtin_nontemporal_store(ev2, (f4*)(e + 2 * H_));
    __builtin_nontemporal_store(ev3, (f4*)(e + 3 * H_));
  }
  if (tid < 4) {  // ev_b for this block's 4 rows
    float e = 0.0f;
    for (int t = 0; t < T_; ++t) e = sF[tid * T_ + t] * e + sD[tid * T_ + t];
    EVB[row0 + tid] = e;
  }
}

// ------------------------------------------------------------------ launch ----
extern "C" void kernel_launch(void* const* d_in, const int* in_sizes, int n_in,
                              void* d_out, int out_size, void* d_ws, size_t ws_size,
                              hipStream_t stream) {
  const float* X   = (const float*)d_in[0];
  const float* H0  = (const float*)d_in[1];
  const float* C0  = (const float*)d_in[2];
  const float* Wih = (const float*)d_in[3];
  const float* Whh = (const float*)d_in[4];
  const float* bih = (const float*)d_in[5];
  const float* bhh = (const float*)d_in[6];

  float* out  = (float*)d_out;
  float* OUT  = out;                                   // [T,B,H]
  float* CX   = OUT  + (size_t)T_ * B_ * H_;           // [B,H]
  float* EVIH = CX   + (size_t)B_ * H_;                // [B,3H,I]
  float* EVHH = EVIH + (size_t)B_ * G3_ * I_;          // [B,3H,H]
  float* EVB  = EVHH + (size_t)B_ * G3_ * H_;          // [B,3H]

  float* ws    = (float*)d_ws;                         // ~53 MB used
  float* gates = ws;                                   // B*4H
  float* cbuf  = gates + (size_t)B_ * G4_;             // B*H
  float* f_all = cbuf  + (size_t)B_ * H_;              // T*B*H
  float* d_all = f_all + (size_t)T_ * B_ * H_;         // T*B*3H

  init_c_kernel<<<(B_ * H_ + 255) / 256, 256, 0, stream>>>(C0, cbuf);

  for (int t = 0; t < T_; ++t) {
    gates_gemm_kernel<<<64, 256, 0, stream>>>(X, H0, OUT, Wih, Whh, bih, bhh,
                                              gates, t);
    pointwise_kernel<<<(B_ * H_) / 256, 256, 0, stream>>>(gates, cbuf, OUT, CX,
                                                          f_all, d_all, t);
  }

  trace_kernel<<<(B_ * G3_) / 4, 192, 0, stream>>>(X, H0, OUT, f_all, d_all,
                                                   EVIH, EVHH, EVB);
}